// TestGradient_20151986553158
// MI455X (gfx1250) — compile-verified
//
#include <hip/hip_runtime.h>
#include <hip/hip_bf16.h>

// ---------------------------------------------------------------------------
// reference: s = sum_{b,r,c,k} X[b,r,k] W[k,c] G[b,k/4,c/4] ; out = s*s
// Factored: xs[b,k] = sum_r X[b,r,k]
//           s = sum_b sum_kb sum_j xs[b,4kb+j] * sum_cb G[b,kb,cb]*wsum(4kb+j,cb)
//           wsum(k,cb) = sum_{i<4} W[k,4cb+i]
// Stage 1 (dominant, 256MB stream) uses V_WMMA_F32_16X16X4_F32 with an
// all-ones A matrix as a tensor-core column reduction.
// ---------------------------------------------------------------------------

typedef float v2f __attribute__((ext_vector_type(2)));
typedef float v8f __attribute__((ext_vector_type(8)));

#define BATCH 8
#define ROWS  2048
#define WIDTH 4096
#define COLS  4096
#define KB    (WIDTH / 4)   // 1024 gate blocks along k
#define CB    (COLS / 4)    // 1024 gate blocks along c
#define ZCHUNKS 8           // row-chunks in stage 1
#define ROWS_PER_Z (ROWS / ZCHUNKS)   // 256

// ---------------------------------------------------------------------------
// Stage 1: xs partials.  part[z][b][k] = sum over rows [z*256, z*256+256) of X[b,r,k]
// grid = (WIDTH/128, BATCH, ZCHUNKS), block = 256 (8 waves, 16 k-columns each)
// WMMA D = ones(16x4) * B(4x16) + C  => each D column n = sum of B column n.
// B layout per ISA (32-bit, K across lane-halves/components, N = lane%16).
// ---------------------------------------------------------------------------
__global__ __launch_bounds__(256) void xsum_wmma_kernel(
    const float* __restrict__ X, float* __restrict__ part) {
  const int wave = threadIdx.x >> 5;
  const int lane = threadIdx.x & 31;
  const int n    = lane & 15;
  const int half = lane >> 4;
  const int k0   = blockIdx.x * 128 + wave * 16;
  const int b    = blockIdx.y;
  const int z    = blockIdx.z;
  const int r0   = z * ROWS_PER_Z;

  const float* xp = X + ((size_t)b * ROWS + r0) * WIDTH + (k0 + n);
  const size_t ro0 = (size_t)(2 * half) * WIDTH;       // K = 0 or 2
  const size_t ro1 = (size_t)(2 * half + 1) * WIDTH;   // K = 1 or 3

  v2f a; a.x = 1.0f; a.y = 1.0f;   // all-ones A matrix (layout-independent)
  v8f c0 = {};
  v8f c1 = {};

  for (int r = 0; r < ROWS_PER_Z; r += 8) {
    const float* p0 = xp + (size_t)r * WIDTH;
    const float* p1 = p0 + (size_t)4 * WIDTH;
    v2f b0; b0.x = p0[ro0]; b0.y = p0[ro1];
    v2f b1; b1.x = p1[ro0]; b1.y = p1[ro1];
    c0 = __builtin_amdgcn_wmma_f32_16x16x4_f32(false, a, false, b0,
                                               (short)0, c0, false, false);
    c1 = __builtin_amdgcn_wmma_f32_16x16x4_f32(false, a, false, b1,
                                               (short)0, c1, false, false);
  }

  // every D row is identical; lane L (<16) holds column k0 + L in c[0]
  if (lane < 16) {
    part[((size_t)z * BATCH + b) * WIDTH + k0 + n] = c0[0] + c1[0];
  }
}

// ---------------------------------------------------------------------------
// Stage 2: gated contraction.  One block per (kb, b).
// acc = sum_cb G[b,kb,cb] * sum_j xs[b,4kb+j] * (sum_i W[4kb+j,4cb+i])
// W rows stay L2-resident across the 8 batches (64MB << 192MB L2).
// ---------------------------------------------------------------------------
__global__ __launch_bounds__(256) void gate_contract_kernel(
    const float* __restrict__ W, const float* __restrict__ G,
    const float* __restrict__ part, float* __restrict__ blockpart) {
  const int kb  = blockIdx.x;
  const int b   = blockIdx.y;
  const int tid = threadIdx.x;

  __shared__ float xs_s[4];
  __shared__ float red[256];

  if (tid < 4) {
    float v = 0.f;
#pragma unroll
    for (int z = 0; z < ZCHUNKS; ++z)
      v += part[((size_t)z * BATCH + b) * WIDTH + 4 * kb + tid];
    xs_s[tid] = v;
  }
  __syncthreads();

  const float x0 = xs_s[0], x1 = xs_s[1], x2 = xs_s[2], x3 = xs_s[3];
  const float* wp = W + (size_t)(4 * kb) * COLS;
  const float* gp = G + ((size_t)b * KB + kb) * CB;

  float acc = 0.f;
  for (int cb = tid; cb < CB; cb += 256) {
    float4 w0 = *(const float4*)(wp + 0 * COLS + 4 * cb);
    float4 w1 = *(const float4*)(wp + 1 * COLS + 4 * cb);
    float4 w2 = *(const float4*)(wp + 2 * COLS + 4 * cb);
    float4 w3 = *(const float4*)(wp + 3 * COLS + 4 * cb);
    float s0 = (w0.x + w0.y) + (w0.z + w0.w);
    float s1 = (w1.x + w1.y) + (w1.z + w1.w);
    float s2 = (w2.x + w2.y) + (w2.z + w2.w);
    float s3 = (w3.x + w3.y) + (w3.z + w3.w);
    acc += gp[cb] * (x0 * s0 + x1 * s1 + x2 * s2 + x3 * s3);
  }

  red[tid] = acc;
  __syncthreads();
  for (int off = 128; off > 0; off >>= 1) {
    if (tid < off) red[tid] += red[tid + off];
    __syncthreads();
  }
  if (tid == 0) blockpart[(size_t)b * KB + kb] = red[0];
}

// ---------------------------------------------------------------------------
// Stage 3: deterministic final sum of the 8192 block partials, then square.
// ---------------------------------------------------------------------------
__global__ __launch_bounds__(256) void finalize_kernel(
    const float* __restrict__ blockpart, float* __restrict__ out) {
  __shared__ float red[256];
  const int tid = threadIdx.x;
  float acc = 0.f;
  for (int i = tid; i < BATCH * KB; i += 256) acc += blockpart[i];
  red[tid] = acc;
  __syncthreads();
  for (int off = 128; off > 0; off >>= 1) {
    if (tid < off) red[tid] += red[tid + off];
    __syncthreads();
  }
  if (tid == 0) {
    float s = red[0];
    out[0] = s * s;
  }
}

extern "C" void kernel_launch(void* const* d_in, const int* in_sizes, int n_in,
                              void* d_out, int out_size, void* d_ws, size_t ws_size,
                              hipStream_t stream) {
  const float* X = (const float*)d_in[0];   // (8, 2048, 4096) f32
  const float* W = (const float*)d_in[1];   // (4096, 4096) f32
  const float* G = (const float*)d_in[2];   // (8, 1024, 1024) f32
  float* out = (float*)d_out;               // scalar f32

  // workspace layout (fully overwritten every call; no zero-init needed)
  float* part      = (float*)d_ws;                                   // 8*8*4096 f32 = 1 MB
  float* blockpart = part + (size_t)ZCHUNKS * BATCH * WIDTH;         // 8192 f32 = 32 KB

  dim3 g1(WIDTH / 128, BATCH, ZCHUNKS);
  xsum_wmma_kernel<<<g1, 256, 0, stream>>>(X, part);

  dim3 g2(KB, BATCH);
  gate_contract_kernel<<<g2, 256, 0, stream>>>(W, G, part, blockpart);

  finalize_kernel<<<1, 256, 0, stream>>>(blockpart, out);
}